// GCN_model2_bn_44487271252088
// MI455X (gfx1250) — compile-verified
//
#include <hip/hip_runtime.h>
#include <math.h>

typedef float v2f __attribute__((ext_vector_type(2)));
typedef float v8f __attribute__((ext_vector_type(8)));

#define FEAT 128
#define N_CLASSES 16
#define BN_EPS 1e-5f

// ---------------------------------------------------------------------------
// utility: fill buffer with constant
// ---------------------------------------------------------------------------
__global__ void k_fill(float* __restrict__ p, float v, int n) {
    int i = blockIdx.x * blockDim.x + threadIdx.x;
    if (i < n) p[i] = v;
}

// ---------------------------------------------------------------------------
// degree accumulation: deg[dst[e]] += 1  (deg pre-filled with 1.0 = self loop)
// ---------------------------------------------------------------------------
__global__ void k_deg_accum(const int* __restrict__ dst, float* __restrict__ deg, int E) {
    int e = blockIdx.x * blockDim.x + threadIdx.x;
    if (e < E) {
        __hip_atomic_fetch_add(&deg[dst[e]], 1.0f,
                               __ATOMIC_RELAXED, __HIP_MEMORY_SCOPE_AGENT);
    }
}

// deg -> deg^{-1/2} in place
__global__ void k_rsqrt_inplace(float* __restrict__ p, int n) {
    int i = blockIdx.x * blockDim.x + threadIdx.x;
    if (i < n) p[i] = rsqrtf(p[i]);
}

// ---------------------------------------------------------------------------
// f32 WMMA GEMM:  C[M,N] = A[M,K] @ B[K,N] (+ bias[N] if bias != null)
// One wave per 16x16 output tile; K consumed in steps of 4 via
// V_WMMA_F32_16X16X4_F32 (full f32 precision matrix pipe).
//
// VGPR layouts (ISA 7.12.2):
//   A 16x4 :  lanes 0-15 -> M=lane, VGPR0={K0|K2 by half}, VGPR1={K1|K3}
//   B 4x16 :  lanes hold N, VGPR0={K0|K2 by half}, VGPR1={K1|K3}
//   C 16x16:  VGPR j -> M = j (lanes 0-15) / j+8 (lanes 16-31), N = lane&15
// ---------------------------------------------------------------------------
__global__ void k_gemm_wmma(const float* __restrict__ A, const float* __restrict__ B,
                            const float* __restrict__ bias, float* __restrict__ C,
                            int M, int K, int N) {
    const int lane  = threadIdx.x & 31;
    const int wave  = threadIdx.x >> 5;
    const int wpb   = blockDim.x >> 5;
    const int tile  = blockIdx.x * wpb + wave;
    const int Mt    = (M + 15) >> 4;
    const int Nt    = N >> 4;
    if (tile >= Mt * Nt) return;

    const int mBase = (tile / Nt) << 4;
    const int nBase = (tile % Nt) << 4;

    int row = mBase + (lane & 15);
    if (row > M - 1) row = M - 1;            // clamp (EXEC must stay all-ones for WMMA)
    const int col  = nBase + (lane & 15);
    const int kOff = (lane >> 4) << 1;       // lanes 0-15: K+0/K+1, lanes 16-31: K+2/K+3

    v8f acc = {};
    for (int k = 0; k < K; k += 4) {
        v2f a = *(const v2f*)(A + (size_t)row * K + k + kOff);
        v2f b;
        b.x = B[(size_t)(k + kOff)     * N + col];
        b.y = B[(size_t)(k + kOff + 1) * N + col];
        acc = __builtin_amdgcn_wmma_f32_16x16x4_f32(
                  /*neg_a=*/false, a, /*neg_b=*/false, b,
                  /*c_mod=*/(short)0, acc, /*reuse_a=*/false, /*reuse_b=*/false);
    }

    const float bb   = bias ? bias[col] : 0.0f;
    const int   rOff = (lane >> 4) << 3;     // 0 or 8
#pragma unroll
    for (int j = 0; j < 8; ++j) {
        int r = mBase + j + rOff;
        if (r < M) C[(size_t)r * N + col] = acc[j] + bb;
    }
}

// ---------------------------------------------------------------------------
// edge aggregation: out[dst] += H[src] * dinv[src]*dinv[dst]
// one wave per edge, each lane owns a float4 slice of the 128 features
// ---------------------------------------------------------------------------
__global__ void k_scatter(const float* __restrict__ H,
                          const int* __restrict__ src, const int* __restrict__ dst,
                          const float* __restrict__ dinv, float* __restrict__ out, int E) {
    const int lane = threadIdx.x & 31;
    const int wave = threadIdx.x >> 5;
    const int e    = blockIdx.x * (blockDim.x >> 5) + wave;
    if (e >= E) return;

    const int   s = src[e];
    const int   d = dst[e];
    const float w = dinv[s] * dinv[d];

    const float4 hv = *(const float4*)(H + (size_t)s * FEAT + lane * 4);
    float* o = out + (size_t)d * FEAT + lane * 4;
    __hip_atomic_fetch_add(o + 0, hv.x * w, __ATOMIC_RELAXED, __HIP_MEMORY_SCOPE_AGENT);
    __hip_atomic_fetch_add(o + 1, hv.y * w, __ATOMIC_RELAXED, __HIP_MEMORY_SCOPE_AGENT);
    __hip_atomic_fetch_add(o + 2, hv.z * w, __ATOMIC_RELAXED, __HIP_MEMORY_SCOPE_AGENT);
    __hip_atomic_fetch_add(o + 3, hv.w * w, __ATOMIC_RELAXED, __HIP_MEMORY_SCOPE_AGENT);
}

// ---------------------------------------------------------------------------
// layer-1 epilogue: self-loop + bias + BatchNorm (running stats) + ReLU
// ---------------------------------------------------------------------------
__global__ void k_post1(float* __restrict__ agg, const float* __restrict__ H,
                        const float* __restrict__ dinv, const float* __restrict__ b1,
                        const float* __restrict__ gamma, const float* __restrict__ beta,
                        const float* __restrict__ mean, const float* __restrict__ var,
                        int total) {
    int i = blockIdx.x * blockDim.x + threadIdx.x;
    if (i >= total) return;
    const int   node = i >> 7;          // /128
    const int   f    = i & (FEAT - 1);
    const float di   = dinv[node];
    float v = agg[i] + H[i] * di * di + b1[f];
    v = (v - mean[f]) * rsqrtf(var[f] + BN_EPS) * gamma[f] + beta[f];
    agg[i] = fmaxf(v, 0.0f);
}

// layer-2 epilogue: self-loop + bias
__global__ void k_post2(float* __restrict__ agg, const float* __restrict__ H,
                        const float* __restrict__ dinv, const float* __restrict__ b2,
                        int total) {
    int i = blockIdx.x * blockDim.x + threadIdx.x;
    if (i >= total) return;
    const int   node = i >> 7;
    const int   f    = i & (FEAT - 1);
    const float di   = dinv[node];
    agg[i] = agg[i] + H[i] * di * di + b2[f];
}

// ---------------------------------------------------------------------------
extern "C" void kernel_launch(void* const* d_in, const int* in_sizes, int n_in,
                              void* d_out, int out_size, void* d_ws, size_t ws_size,
                              hipStream_t stream) {
    const float* X     = (const float*)d_in[0];
    const int*   edge  = (const int*)  d_in[1];
    const float* W1    = (const float*)d_in[2];
    const float* b1    = (const float*)d_in[3];
    const float* gamma = (const float*)d_in[4];
    const float* beta  = (const float*)d_in[5];
    const float* rmean = (const float*)d_in[6];
    const float* rvar  = (const float*)d_in[7];
    const float* W2    = (const float*)d_in[8];
    const float* b2    = (const float*)d_in[9];
    const float* Wc    = (const float*)d_in[10];
    const float* bc    = (const float*)d_in[11];
    float*       out   = (float*)d_out;

    const int N = in_sizes[0] / FEAT;       // 50000 nodes
    const int E = in_sizes[1] / 2;          // 640000 edges
    const int* src = edge;                  // edge_index[0]
    const int* dst = edge + E;              // edge_index[1]
    const int total = N * FEAT;

    // workspace layout (~52 MB): deg/dinv | bufA (H pre-agg) | bufB (aggregated)
    float* deg  = (float*)d_ws;
    float* bufA = deg + ((N + 63) & ~63);
    float* bufB = bufA + (size_t)total;

    const int T256   = 256;
    const int gNode  = (N + T256 - 1) / T256;
    const int gEdge  = (E + T256 - 1) / T256;
    const int gFeat  = (total + T256 - 1) / T256;
    const int gScat  = (E + 7) / 8;                         // 8 waves (edges) / block
    const int tilesL = ((N + 15) / 16) * (FEAT / 16);       // 16x16 tiles, layer GEMMs
    const int tilesC = ((N + 15) / 16) * (N_CLASSES / 16);  // classifier tiles
    const int gGemmL = (tilesL + 7) / 8;                    // 8 waves (tiles) / block
    const int gGemmC = (tilesC + 7) / 8;

    // --- symmetric normalization: dinv = rsqrt(deg + selfloop) ---
    k_fill<<<gNode, T256, 0, stream>>>(deg, 1.0f, N);
    k_deg_accum<<<gEdge, T256, 0, stream>>>(dst, deg, E);
    k_rsqrt_inplace<<<gNode, T256, 0, stream>>>(deg, N);

    // --- layer 1: H = X @ W1 ; aggregate ; +b1, BN, ReLU ---
    k_gemm_wmma<<<gGemmL, T256, 0, stream>>>(X, W1, nullptr, bufA, N, FEAT, FEAT);
    k_fill<<<gFeat, T256, 0, stream>>>(bufB, 0.0f, total);
    k_scatter<<<gScat, T256, 0, stream>>>(bufA, src, dst, deg, bufB, E);
    k_post1<<<gFeat, T256, 0, stream>>>(bufB, bufA, deg, b1, gamma, beta, rmean, rvar, total);

    // --- layer 2: H = act @ W2 ; aggregate ; +b2 ---
    k_gemm_wmma<<<gGemmL, T256, 0, stream>>>(bufB, W2, nullptr, bufA, N, FEAT, FEAT);
    k_fill<<<gFeat, T256, 0, stream>>>(bufB, 0.0f, total);
    k_scatter<<<gScat, T256, 0, stream>>>(bufA, src, dst, deg, bufB, E);
    k_post2<<<gFeat, T256, 0, stream>>>(bufB, bufA, deg, b2, total);

    // --- classifier: out = act @ Wc + bc ---
    k_gemm_wmma<<<gGemmC, T256, 0, stream>>>(bufB, Wc, bc, out, N, FEAT, N_CLASSES);
}